// SGConv_9079560864435
// MI455X (gfx1250) — compile-verified
//
#include <hip/hip_runtime.h>
#include <hip/hip_bf16.h>
#include <math.h>

// Problem constants (match reference)
#define BATCH 128
#define SEQ   64
#define FIN   32
#define FOUT  32
#define NNODE 2048   // SEQ*FIN

typedef float v2f __attribute__((ext_vector_type(2)));
typedef float v8f __attribute__((ext_vector_type(8)));

#define GLOBAL_AS __attribute__((address_space(1)))
#define LDS_AS    __attribute__((address_space(3)))

// The async builtins take typed b128 (= int4 vector) pointers:
//   param0: global (addrspace(1)) int4*,  param1: LDS (addrspace(3)) int4*
typedef int b128_t __attribute__((vector_size(16)));
typedef GLOBAL_AS b128_t* gb128p;
typedef LDS_AS    b128_t* lb128p;

// gfx1250 async copy paths (guarded: fall back to VGPR copies if the
// toolchain doesn't declare the builtins)
#if defined(__AMDGCN__) && __has_builtin(__builtin_amdgcn_global_load_async_to_lds_b128)
#define HAVE_ASYNC_LD 1
#else
#define HAVE_ASYNC_LD 0
#endif
#if defined(__AMDGCN__) && __has_builtin(__builtin_amdgcn_global_store_async_from_lds_b128)
#define HAVE_ASYNC_ST 1
#else
#define HAVE_ASYNC_ST 0
#endif

__device__ __forceinline__ void async_copy_b128(const float* gsrc, float* ldst) {
#if HAVE_ASYNC_LD
  __builtin_amdgcn_global_load_async_to_lds_b128((gb128p)gsrc, (lb128p)ldst, 0, 0);
#else
  *(float4*)ldst = *(const float4*)gsrc;
#endif
}

__device__ __forceinline__ void wait_async_loads() {
#if HAVE_ASYNC_LD
#if __has_builtin(__builtin_amdgcn_s_wait_asynccnt)
  __builtin_amdgcn_s_wait_asynccnt(0);
#else
  asm volatile("s_wait_asynccnt 0x0" ::: "memory");
#endif
  asm volatile("" ::: "memory");  // keep DS reads after the wait
#endif
}

// ---------------------------------------------------------------------------
// Kernel A: build the 2 distinct Laplacian diagonal blocks + 2 distinct
// off-diagonal coefficient vectors, estimate ||L||_2 by power iteration on
// L^T L (L is block-tridiagonal, 2048x2048 logically, tiny physically),
// then write the *scaled* blocks of L_scaled = 2L/norm - I to workspace.
//
// ws layout (floats):
//   [   0..1023] Ds_edge (32x32)   diag block for s==0 or s==63
//   [1024..2047] Ds_int  (32x32)   diag block for 1<=s<=62
//   [2048..2079] cs_edge (32)      off-diag coeff for edge q in {0,62}
//   [2080..2111] cs_int  (32)      off-diag coeff for 1<=q<=61
// ---------------------------------------------------------------------------
__global__ __launch_bounds__(256) void cheb_setup_kernel(
    const float* __restrict__ adj, float* __restrict__ ws) {
  __shared__ float blk[32 * 32];
  __shared__ float Dedge[32 * 32];
  __shared__ float Dint[32 * 32];
  __shared__ float disE[32], disI[32], cE[32], cI[32];
  __shared__ float v[NNODE];
  __shared__ float y[NNODE];
  __shared__ float red[256];

  const int tid = threadIdx.x;

  // feature block: sigmoid(adj_param), zero diagonal
  for (int i = tid; i < 1024; i += 256) {
    int f = i >> 5, g = i & 31;
    float a = adj[i];
    float sg = 1.0f / (1.0f + __expf(-a));
    blk[i] = (f == g) ? 0.0f : sg;
  }
  __syncthreads();

  // degrees: d = self(1) + temporal(1 or 2) + feature row-sum
  if (tid < 32) {
    float rs = 0.0f;
    for (int g = 0; g < 32; ++g) rs += blk[tid * 32 + g];
    disE[tid] = rsqrtf(2.0f + rs);  // boundary timestep (one temporal edge)
    disI[tid] = rsqrtf(3.0f + rs);  // interior timestep (two temporal edges)
  }
  __syncthreads();

  // L diagonal blocks: delta - dis_f * (blk + delta) * dis_g
  for (int i = tid; i < 1024; i += 256) {
    int f = i >> 5, g = i & 31;
    float aij = blk[i] + ((f == g) ? 1.0f : 0.0f);
    float dlt = (f == g) ? 1.0f : 0.0f;
    Dedge[i] = dlt - disE[f] * aij * disE[g];
    Dint[i]  = dlt - disI[f] * aij * disI[g];
  }
  if (tid < 32) {
    cE[tid] = -disE[tid] * disI[tid];  // off-diag L entries (symmetric)
    cI[tid] = -disI[tid] * disI[tid];
  }
  for (int i = tid; i < NNODE; i += 256) v[i] = 1.0f + 0.01f * (float)(i & 7);
  __syncthreads();

  // power iteration on M = L^T L  ->  sigma_max(L) = sqrt(lambda_max(M))
  float lam = 1.0f;
  for (int it = 0; it < 128; ++it) {
    // y = L v   (row block s: Dblk(s)*v_s + c(s-1).*v_{s-1} + c(s).*v_{s+1})
    for (int node = tid; node < NNODE; node += 256) {
      int sb = node >> 5, f = node & 31;
      const float* D = (sb == 0 || sb == 63) ? Dedge : Dint;
      const float* vs = &v[sb << 5];
      float dot = 0.0f;
      for (int g = 0; g < 32; ++g) dot += D[f * 32 + g] * vs[g];
      if (sb > 0)  dot += (((sb - 1) == 0 || sb == 63) ? cE[f] : cI[f]) * v[node - 32];
      if (sb < 63) dot += ((sb == 0 || sb == 62) ? cE[f] : cI[f]) * v[node + 32];
      y[node] = dot;
    }
    __syncthreads();
    // w = L^T y  (transpose uses D[g][f]; off-diag coeffs are symmetric)
    float part = 0.0f;
    float wloc[8];
#pragma unroll
    for (int t = 0; t < 8; ++t) {
      int node = tid + t * 256;
      int sb = node >> 5, f = node & 31;
      const float* D = (sb == 0 || sb == 63) ? Dedge : Dint;
      const float* ys = &y[sb << 5];
      float dot = 0.0f;
      for (int g = 0; g < 32; ++g) dot += D[g * 32 + f] * ys[g];
      if (sb > 0)  dot += (((sb - 1) == 0 || sb == 63) ? cE[f] : cI[f]) * y[node - 32];
      if (sb < 63) dot += ((sb == 0 || sb == 62) ? cE[f] : cI[f]) * y[node + 32];
      wloc[t] = dot;
      part += dot * dot;
    }
    red[tid] = part;
    __syncthreads();
    for (int off = 128; off > 0; off >>= 1) {
      if (tid < off) red[tid] += red[tid + off];
      __syncthreads();
    }
    float nrm = sqrtf(red[0]);
    lam = nrm;  // ||Mv|| -> lambda_max as v converges (v was unit)
    float inv = (nrm > 0.0f) ? 1.0f / nrm : 0.0f;
#pragma unroll
    for (int t = 0; t < 8; ++t) v[tid + t * 256] = wloc[t] * inv;
    __syncthreads();
  }

  float sigma = sqrtf(lam);
  float nrm = (sigma > 1e-6f) ? sigma : 1.0f;
  float scale = 2.0f / nrm;

  // L_scaled blocks: scale*D - I  and scale*c
  for (int i = tid; i < 1024; i += 256) {
    int f = i >> 5, g = i & 31;
    float dlt = (f == g) ? 1.0f : 0.0f;
    ws[i]        = scale * Dedge[i] - dlt;
    ws[1024 + i] = scale * Dint[i]  - dlt;
  }
  if (tid < 32) {
    ws[2048 + tid] = scale * cE[tid];
    ws[2080 + tid] = scale * cI[tid];
  }
}

// ---------------------------------------------------------------------------
// WMMA helpers: fp32 16x16x4, chained over K=32, two 16-column halves.
// A (16x32) row-major in LDS, B (32x32) row-major in LDS.
// A frag layout: lanes 0-15 M=lane hold K=kk+{0,1}; lanes 16-31 K=kk+{2,3}.
// B frag layout: v0/v1 hold rows K=kk+kg / kk+kg+1 at column N=lane&15.
// C/D layout: vgpr r, lane l -> M = r + 8*(l>>4), N = l&15 (per 16-col half).
// ---------------------------------------------------------------------------
__device__ __forceinline__ void gemm16x32_half(
    const float* __restrict__ A, const float* __restrict__ B,
    int nh, int lane, v8f& acc) {
  const int m  = lane & 15;
  const int kg = (lane >> 4) << 1;  // 0 or 2
  const int n  = nh * 16 + (lane & 15);
#pragma unroll
  for (int kk = 0; kk < 32; kk += 4) {
    v2f a, b;
    a.x = A[m * 32 + kk + kg];
    a.y = A[m * 32 + kk + kg + 1];
    b.x = B[(kk + kg) * 32 + n];
    b.y = B[(kk + kg + 1) * 32 + n];
    acc = __builtin_amdgcn_wmma_f32_16x16x4_f32(
        false, a, false, b, (short)0, acc, false, false);
  }
}

__device__ __forceinline__ void add_scaled(
    const float* __restrict__ X, const float* __restrict__ c,
    int nh, int lane, v8f& acc) {
  const int hi = lane >> 4;
  const int n  = nh * 16 + (lane & 15);
  const float cf = c[n];
#pragma unroll
  for (int r = 0; r < 8; ++r) acc[r] += cf * X[(r + 8 * hi) * 32 + n];
}

__device__ __forceinline__ void store_tile(
    float* __restrict__ Z, int nh, int lane, const v8f& acc) {
  const int hi = lane >> 4;
  const int n  = nh * 16 + (lane & 15);
#pragma unroll
  for (int r = 0; r < 8; ++r) Z[(r + 8 * hi) * 32 + n] = acc[r];
}

__device__ __forceinline__ float fixup(float v) {
  if (v != v) return 0.0f;
  if (v == __builtin_inff()) return 1e6f;
  if (v == -__builtin_inff()) return -1e6f;
  return v;
}

// ---------------------------------------------------------------------------
// Kernel B: one wave (32 lanes) per (batch-tile of 16, timestep s).
// z1 = x @ Ls  (block-banded), z2 = 2*z1 @ Ls - x,
// out = x@W0 + z1@W1 + z2@W2 + bias.   7 WMMA GEMMs of 16x32 @ 32x32.
// Single wave => LDS ops are in-order, no barriers; branches are uniform
// so EXEC is all-ones around every WMMA. Operand staging uses the gfx1250
// async global->LDS engine (ASYNCcnt) when available.
// ---------------------------------------------------------------------------
__global__ __launch_bounds__(32) void sgconv_kernel(
    const float* __restrict__ x, const float* __restrict__ weight,
    const float* __restrict__ bias, const float* __restrict__ ws,
    float* __restrict__ out) {
  const int s    = blockIdx.x;        // 0..63
  const int b0   = blockIdx.y * 16;   // batch tile base
  const int lane = threadIdx.x;

  __shared__ __align__(16) float xb[5][16 * 32];   // x block-cols s-2..s+2
  __shared__ __align__(16) float z1[3][16 * 32];   // z1 block-cols s-1..s+1
  __shared__ __align__(16) float z2t[16 * 32];
  __shared__ __align__(16) float Dsh[3][32 * 32];  // Ls diag blocks s-1..s+1
  __shared__ __align__(16) float Wsh[3][32 * 32];  // W0..W2
  __shared__ __align__(16) float obuf[16 * 32];    // output bounce buffer
  __shared__ float csh[4][32];                     // cs(s-2..s+1)
  __shared__ float bsh[32];

  __builtin_prefetch(weight, 0, 3);
  __builtin_prefetch(ws, 0, 3);

  const float4 fzero = {0.0f, 0.0f, 0.0f, 0.0f};

  // stage x tiles: 4 chunks of b128 per 16x32 tile (lane*16B each)
#pragma unroll
  for (int t = 0; t < 5; ++t) {
    int jj = s - 2 + t;
    if (jj >= 0 && jj < SEQ) {
#pragma unroll
      for (int c = 0; c < 4; ++c) {
        int i = c * 128 + lane * 4;
        int m = i >> 5, f = i & 31;
        async_copy_b128(&x[((b0 + m) * SEQ + jj) * FIN + f], &xb[t][i]);
      }
    } else {
#pragma unroll
      for (int c = 0; c < 4; ++c) *(float4*)&xb[t][c * 128 + lane * 4] = fzero;
    }
  }
  // stage Ls diag blocks + weights: 8 chunks of b128 per 32x32 matrix
#pragma unroll
  for (int t = 0; t < 3; ++t) {
    int jj = s - 1 + t;
    if (jj >= 0 && jj < SEQ) {
      const float* Dsrc = ws + (((jj == 0) || (jj == SEQ - 1)) ? 0 : 1024);
#pragma unroll
      for (int c = 0; c < 8; ++c) {
        int i = c * 128 + lane * 4;
        async_copy_b128(Dsrc + i, &Dsh[t][i]);
      }
    } else {
#pragma unroll
      for (int c = 0; c < 8; ++c) *(float4*)&Dsh[t][c * 128 + lane * 4] = fzero;
    }
#pragma unroll
    for (int c = 0; c < 8; ++c) {
      int i = c * 128 + lane * 4;
      async_copy_b128(weight + t * 1024 + i, &Wsh[t][i]);
    }
  }
  // stage off-diag coefficient vectors + bias (tiny: plain path)
#pragma unroll
  for (int t = 0; t < 4; ++t) {
    int q = s - 2 + t;  // edge between block-cols q and q+1
    const float* csrc = ws + (((q == 0) || (q == SEQ - 2)) ? 2048 : 2080);
    csh[t][lane] = (q >= 0 && q < SEQ - 1) ? csrc[lane] : 0.0f;
  }
  bsh[lane] = bias[lane];

  wait_async_loads();  // s_wait_asynccnt 0: LDS tiles are now valid

  // z1 block-cols j = s-1..s+1 :  z1_j = x_{j-1}.*c(j-1) + x_j@Ds(j) + x_{j+1}.*c(j)
#pragma unroll
  for (int lj = 0; lj < 3; ++lj) {
    int jj = s - 1 + lj;
    if (jj < 0 || jj >= SEQ) {
      for (int i = lane; i < 512; i += 32) z1[lj][i] = 0.0f;
      continue;
    }
#pragma unroll
    for (int nh = 0; nh < 2; ++nh) {
      v8f acc = {};
      gemm16x32_half(xb[lj + 1], Dsh[lj], nh, lane, acc);
      if (jj > 0)       add_scaled(xb[lj],     csh[lj],     nh, lane, acc);
      if (jj < SEQ - 1) add_scaled(xb[lj + 2], csh[lj + 1], nh, lane, acc);
      store_tile(z1[lj], nh, lane, acc);
    }
  }

  // z2_s = 2*(z1_{s-1}.*c(s-1) + z1_s@Ds(s) + z1_{s+1}.*c(s)) - x_s
#pragma unroll
  for (int nh = 0; nh < 2; ++nh) {
    v8f acc = {};
    gemm16x32_half(z1[1], Dsh[1], nh, lane, acc);
    if (s > 0)       add_scaled(z1[0], csh[1], nh, lane, acc);
    if (s < SEQ - 1) add_scaled(z1[2], csh[2], nh, lane, acc);
    const int hi = lane >> 4, n = nh * 16 + (lane & 15);
#pragma unroll
    for (int r = 0; r < 8; ++r)
      acc[r] = 2.0f * acc[r] - xb[2][(r + 8 * hi) * 32 + n];
    store_tile(z2t, nh, lane, acc);
  }

  // out_s = x_s@W0 + z1_s@W1 + z2_s@W2 + bias
#pragma unroll
  for (int nh = 0; nh < 2; ++nh) {
    v8f acc = {};
    gemm16x32_half(xb[2], Wsh[0], nh, lane, acc);
    gemm16x32_half(z1[1], Wsh[1], nh, lane, acc);
    gemm16x32_half(z2t,  Wsh[2], nh, lane, acc);
    const int hi = lane >> 4, n = nh * 16 + (lane & 15);
    const float bv = bsh[n];
#pragma unroll
    for (int r = 0; r < 8; ++r) {
      int m = r + 8 * hi;
      float vv = fixup(acc[r] + bv);
#if HAVE_ASYNC_ST
      obuf[m * 32 + n] = vv;
#else
      out[((b0 + m) * SEQ + s) * FOUT + n] = vv;
#endif
    }
  }

#if HAVE_ASYNC_ST
  // drain DS writes to obuf, then stream it out on the async engine;
  // s_endpgm performs an implicit wait-idle so the stores complete.
  asm volatile("s_wait_dscnt 0x0" ::: "memory");
#pragma unroll
  for (int c = 0; c < 4; ++c) {
    int i = c * 128 + lane * 4;
    int m = i >> 5, f = i & 31;
    __builtin_amdgcn_global_store_async_from_lds_b128(
        (gb128p)(out + ((b0 + m) * SEQ + s) * FOUT + f),
        (lb128p)(&obuf[i]), 0, 0);
  }
#else
  (void)obuf;
#endif
}

// ---------------------------------------------------------------------------
extern "C" void kernel_launch(void* const* d_in, const int* in_sizes, int n_in,
                              void* d_out, int out_size, void* d_ws, size_t ws_size,
                              hipStream_t stream) {
  (void)in_sizes; (void)n_in; (void)out_size; (void)ws_size;
  const float* x      = (const float*)d_in[0];   // [128,64,32]
  const float* weight = (const float*)d_in[1];   // [3,32,32]
  const float* bias   = (const float*)d_in[2];   // [32]
  const float* adj    = (const float*)d_in[3];   // [32,32]
  float* out = (float*)d_out;                    // [128,64,32]
  float* ws  = (float*)d_ws;                     // >= 2112 floats used

  cheb_setup_kernel<<<1, 256, 0, stream>>>(adj, ws);
  sgconv_kernel<<<dim3(SEQ, BATCH / 16), 32, 0, stream>>>(x, weight, bias, ws, out);
}